// TopkRouting_38130719654000
// MI455X (gfx1250) — compile-verified
//
#include <hip/hip_runtime.h>
#include <stdint.h>

// TopkRouting: per row of M=2048 floats, top-64 of |x| (desc, ties->lower idx),
// per-sample softmax over first k_n entries, zero padding.
// Memory-bound: 134MB in -> ~5.8us floor @ 23.3 TB/s. Algorithm is O(M)/row:
// 4x 8-bit radix-histogram select -> exact 64th key -> compact -> rank-sort -> softmax.
// CDNA5 paths: TDM tensor_load_to_lds (global->LDS DMA, TENSORcnt) + wave32 shuffles.

#define M_DIM   2048
#define KSEL    64
#define BLOCK   256
#define NWAVE   (BLOCK / 32)
#define EPT     (M_DIM / BLOCK)      // 8 elements per thread
#define HSTRIDE 264                  // 256 bins + 8 pad: per-wave copies hit distinct banks

typedef unsigned int uint32;
typedef unsigned int v4u __attribute__((ext_vector_type(4)));
typedef int          v8i __attribute__((ext_vector_type(8)));
typedef int          v4i __attribute__((ext_vector_type(4)));

__global__ __launch_bounds__(BLOCK) void topk_routing_kernel(
    const float* __restrict__ gen_adj,
    const long long* __restrict__ topk,   // int64 per reference
    float* __restrict__ out_w,
    float* __restrict__ out_i,
    int Q)
{
  __shared__ float  rowv[M_DIM];                  // 8KB TDM destination
  __shared__ uint32 histp[NWAVE * HSTRIDE];       // per-wave privatized histograms
  __shared__ uint32 selKey[KSEL], selIdx[KSEL];
  __shared__ uint32 srtKey[KSEL], srtIdx[KSEL];
  __shared__ float  sExp[KSEL];
  __shared__ uint32 s_bd, s_bg, s_ctr[2];

  const int tid  = (int)threadIdx.x;
  const int lane = tid & 31;                      // wave32 (gfx1250)
  const int wv   = tid >> 5;
  const int row  = (int)blockIdx.x;
  const float* g_row = gen_adj + (size_t)row * M_DIM;

  uint32 key[EPT];

#if __has_builtin(__builtin_amdgcn_tensor_load_to_lds) && __has_builtin(__builtin_amdgcn_s_wait_tensorcnt)
  {
    // Each of the 8 waves DMAs its own 256-float (1KB) chunk of the row into LDS
    // via the Tensor Data Mover. Descriptor per ISA ch.8:
    //   group0: count=1, lds_addr, global_addr[56:0], type=2
    //   group1: data_size=2 (4B), tensor_dim0=tile_dim0=256, tensor_dim1=tile_dim1=1,
    //           tensor_dim0_stride=256, no padding/iterate/multicast
    const uint32 wave = (uint32)__builtin_amdgcn_readfirstlane((int)((unsigned)tid >> 5));
    const uint32 ce = M_DIM / NWAVE;              // 256 elements per chunk
    unsigned long long ga = (unsigned long long)(uintptr_t)(g_row + wave * ce);
    const uint32 la = (uint32)(uintptr_t)(void*)&rowv[wave * ce];   // low 32 bits = LDS offset
    v4u g0 = { 1u,                                                  // [1:0] count=1
               la,                                                  // [63:32] lds_addr
               (uint32)ga,                                          // [95:64] global_addr lo
               (uint32)(((ga >> 32) & 0x01FFFFFFu) | 0x80000000u) };// [120:96] ga hi, [127:126]=2
    v8i g1 = { (int)0x00020000u,        // data_size=4B
               (int)(ce << 16),         // tensor_dim0[15:0] @ bits[63:48]
               (int)(1u << 16),         // tensor_dim1=1     @ bits[95:80]
               (int)(ce << 16),         // tile_dim0=256     @ bits[127:112]
               1,                       // tile_dim1=1
               (int)ce,                 // tensor_dim0_stride lo32
               0, 0 };
    v4i gz4 = { 0, 0, 0, 0 };           // groups 2/3 unused (2D tensor)
    v8i gz8 = { 0, 0, 0, 0, 0, 0, 0, 0 };
    // amdgpu-toolchain (clang-23) 6-arg form: (v4u, v8i, v4i, v4i, v8i, i32 cpol)
    __builtin_amdgcn_tensor_load_to_lds(g0, g1, gz4, gz4, gz8, 0);
    __builtin_amdgcn_s_wait_tensorcnt(0);
  }
  __syncthreads();                      // make all chunks visible to all waves
#pragma unroll
  for (int j = 0; j < EPT; ++j)         // bank-conflict-free: stride-1 across lanes
    key[j] = __float_as_uint(rowv[j * BLOCK + tid]) & 0x7FFFFFFFu;  // bits of |x|
#else
#pragma unroll
  for (int j = 0; j < EPT; ++j)
    key[j] = __float_as_uint(g_row[j * BLOCK + tid]) & 0x7FFFFFFFu;
#endif

  // ---- 4-pass radix select: find exact key T of the 64th-largest element ----
  uint32 prefix = 0, pmask = 0, remaining = KSEL;
  for (int pass = 0; pass < 4; ++pass) {
    const int shift = 24 - 8 * pass;
    for (int i = tid; i < NWAVE * HSTRIDE; i += BLOCK) histp[i] = 0;
    __syncthreads();
#pragma unroll
    for (int j = 0; j < EPT; ++j) {
      const uint32 k = key[j];
      if ((k & pmask) == prefix)
        atomicAdd(&histp[wv * HSTRIDE + ((k >> shift) & 255u)], 1u);
    }
    __syncthreads();
    if (tid < 32) {
      // wave 0: merge 8 copies + suffix-scan 256 bins with shuffles (no barriers)
      const int g = 31 - lane;                     // lane 0 owns the top 8 bins
      uint32 hloc[8];
#pragma unroll
      for (int b = 0; b < 8; ++b) hloc[b] = 0;
#pragma unroll
      for (int c = 0; c < NWAVE; ++c)
#pragma unroll
        for (int b = 0; b < 8; ++b)
          hloc[b] += histp[c * HSTRIDE + g * 8 + b];
      uint32 s = 0;
#pragma unroll
      for (int b = 0; b < 8; ++b) s += hloc[b];
      const uint32 sorig = s;
#pragma unroll
      for (int off = 1; off < 32; off <<= 1) {     // inclusive scan over lanes (wave32)
        const uint32 t = __shfl_up(s, off);
        if (lane >= off) s += t;
      }
      const uint32 excl = s - sorig;               // count in strictly-higher bin groups
      if (excl < remaining && excl + sorig >= remaining) {  // exactly one lane hits
        uint32 cum = excl, d = (uint32)(g * 8);
        for (int b = 7; b >= 0; --b) {
          if (cum + hloc[b] >= remaining) { d = (uint32)(g * 8 + b); break; }
          cum += hloc[b];
        }
        s_bd = d; s_bg = cum;                      // cum = #elements strictly above bin d
      }
    }
    __syncthreads();
    const uint32 d = s_bd;
    remaining -= s_bg;
    prefix |= d << shift;
    pmask  |= 0xFFu << shift;
  }
  const uint32 T = prefix;                         // exact 64th-largest key
  // remaining = how many key==T elements belong in the top-64

  // ---- compact exactly 64 (key,index) pairs into LDS ----
  if (tid < 2) s_ctr[tid] = 0;
  __syncthreads();
#pragma unroll
  for (int j = 0; j < EPT; ++j) {
    const uint32 k = key[j];
    if (k >= T) {
      const uint32 idx = (uint32)(j * BLOCK + tid);
      if (k > T) {
        const uint32 p = atomicAdd(&s_ctr[0], 1u);
        selKey[p] = k; selIdx[p] = idx;
      } else {
        const uint32 e = atomicAdd(&s_ctr[1], 1u); // capped slots for == T
        if (e < remaining) {
          const uint32 p = atomicAdd(&s_ctr[0], 1u);
          selKey[p] = k; selIdx[p] = idx;
        }
      }
    }
  }
  __syncthreads();

  // ---- rank-sort the 64 (desc by key, ties -> lower index); same-address LDS
  // reads broadcast across the 64 threads, so this is conflict-free ----
  if (tid < KSEL) {
    const uint32 k = selKey[tid], i = selIdx[tid];
    int rank = 0;
    for (int j = 0; j < KSEL; ++j) {
      const uint32 kj = selKey[j], ij = selIdx[j];
      rank += (kj > k || (kj == k && ij < i)) ? 1 : 0;
    }
    srtKey[rank] = k; srtIdx[rank] = i;
  }
  __syncthreads();

  // ---- masked softmax (matches jnp.where(valid, vals, -1e30) semantics) ----
  const int n = row / Q;
  long long kraw = topk[n];
  if (kraw > (long long)M_DIM) kraw = (long long)M_DIM;   // min(topk, M)
  const int ki = (int)kraw;                               // >= 1 per reference

  if (tid < KSEL) {
    const float vmax = __uint_as_float(srtKey[0]);        // rank 0 = row max, always valid
    const float v    = __uint_as_float(srtKey[tid]);
    sExp[tid] = (tid < ki) ? __expf(v - vmax) : 0.0f;     // padded slots contribute 0
  }
  __syncthreads();
  if (tid < KSEL) {
    float sum = 0.0f;
#pragma unroll
    for (int j = 0; j < KSEL; ++j) sum += sExp[j];
    const size_t o = (size_t)row * KSEL + (size_t)tid;
    out_w[o] = (tid < ki) ? sExp[tid] / sum : 0.0f;
    // topk_index is integral in the reference; d_out is float-typed, so emit the
    // numeric value (indices < 2048 are exact in fp32). Padded slots -> 0.
    out_i[o] = (tid < ki) ? (float)srtIdx[tid] : 0.0f;
  }
}

extern "C" void kernel_launch(void* const* d_in, const int* in_sizes, int n_in,
                              void* d_out, int out_size, void* d_ws, size_t ws_size,
                              hipStream_t stream) {
  const float*     gen_adj = (const float*)d_in[0];
  const long long* topk    = (const long long*)d_in[1];   // int64 input
  const int N = in_sizes[1];                              // topk has N elements
  const int Q = in_sizes[0] / (N * M_DIM);
  const int rows = N * Q;                                 // 16384 blocks, 8 waves each
  float* out_w = (float*)d_out;
  float* out_i = out_w + (size_t)rows * KSEL;             // outputs concatenated flat
  topk_routing_kernel<<<dim3(rows), dim3(BLOCK), 0, stream>>>(
      gen_adj, topk, out_w, out_i, Q);
}